// InductiveLaplacianODEFunc_51049981280260
// MI455X (gfx1250) — compile-verified
//
#include <hip/hip_runtime.h>
#include <math.h>

#define N_NODES 50000
#define DD 2
#define HCH 128
#define NE0 200000
#define MROWS (N_NODES * DD)   // 100000
#define ROWLEN (DD * HCH)      // 256

typedef float v2f __attribute__((ext_vector_type(2)));
typedef float v4f __attribute__((ext_vector_type(4)));
typedef float v8f __attribute__((ext_vector_type(8)));

// ---------------------------------------------------------------------------
// Kernel 0: zero the diag accumulator
// ---------------------------------------------------------------------------
__global__ void k_zero(float* __restrict__ p, int n) {
    int i = blockIdx.x * blockDim.x + threadIdx.x;
    if (i < n) p[i] = 0.0f;
}

// ---------------------------------------------------------------------------
// Kernel 1: per-node projections a[n] = x_maps[n] @ Wa.T, b[n] = x_maps[n] @ Wb.T
// One wave (32 lanes) per node; 8 floats per lane; shuffle reduction.
// W_sheaf is (2, 512) row-major: Wa[d',k] = W_sheaf[d'*512 + k], Wb shifted by 256.
// ---------------------------------------------------------------------------
__global__ void k_node_proj(const float* __restrict__ x,
                            const float* __restrict__ W_sheaf,
                            float* __restrict__ a, float* __restrict__ b) {
    __shared__ float sw[1024];
    int tid = threadIdx.x;
    for (int i = tid; i < 1024; i += 256) sw[i] = W_sheaf[i];
    __syncthreads();

    int wave = tid >> 5, lane = tid & 31;
    int n = blockIdx.x * 8 + wave;
    if (n >= N_NODES) return;

    const float* xr = x + (size_t)n * ROWLEN;
    int k0 = lane * 8;
    v4f x0 = *(const v4f*)(xr + k0);
    v4f x1 = *(const v4f*)(xr + k0 + 4);

    float pa0 = 0.f, pa1 = 0.f, pb0 = 0.f, pb1 = 0.f;
#pragma unroll
    for (int j = 0; j < 4; ++j) {
        float xv = x0[j]; int k = k0 + j;
        pa0 += xv * sw[k];
        pa1 += xv * sw[512 + k];
        pb0 += xv * sw[256 + k];
        pb1 += xv * sw[768 + k];
    }
#pragma unroll
    for (int j = 0; j < 4; ++j) {
        float xv = x1[j]; int k = k0 + 4 + j;
        pa0 += xv * sw[k];
        pa1 += xv * sw[512 + k];
        pb0 += xv * sw[256 + k];
        pb1 += xv * sw[768 + k];
    }
#pragma unroll
    for (int off = 16; off > 0; off >>= 1) {
        pa0 += __shfl_down(pa0, off, 32);
        pa1 += __shfl_down(pa1, off, 32);
        pb0 += __shfl_down(pb0, off, 32);
        pb1 += __shfl_down(pb1, off, 32);
    }
    if (lane == 0) {
        a[2 * n + 0] = pa0; a[2 * n + 1] = pa1;
        b[2 * n + 0] = pb0; b[2 * n + 1] = pb1;
    }
}

// ---------------------------------------------------------------------------
// Kernel 2: per-undirected-edge maps, diag accumulation, p = m1*m2 (symmetric)
// ---------------------------------------------------------------------------
__global__ void k_edge_maps(const int* __restrict__ src, const int* __restrict__ dst,
                            const float* __restrict__ a, const float* __restrict__ b,
                            float* __restrict__ diag, float* __restrict__ p) {
    int i = blockIdx.x * blockDim.x + threadIdx.x;
    if (i >= NE0) return;
    int s = src[i], d = dst[i];
    float m1_0 = tanhf(a[2 * s + 0] + b[2 * d + 0]);
    float m1_1 = tanhf(a[2 * s + 1] + b[2 * d + 1]);
    float m2_0 = tanhf(a[2 * d + 0] + b[2 * s + 0]);
    float m2_1 = tanhf(a[2 * d + 1] + b[2 * s + 1]);
    unsafeAtomicAdd(&diag[2 * s + 0], m1_0 * m1_0);
    unsafeAtomicAdd(&diag[2 * s + 1], m1_1 * m1_1);
    unsafeAtomicAdd(&diag[2 * d + 0], m2_0 * m2_0);
    unsafeAtomicAdd(&diag[2 * d + 1], m2_1 * m2_1);
    p[2 * i + 0] = m1_0 * m2_0;
    p[2 * i + 1] = m1_1 * m2_1;
}

// ---------------------------------------------------------------------------
// Kernel 3: dis = rsqrt(diag+1), diag_scaled = diag * dis^2
// ---------------------------------------------------------------------------
__global__ void k_node_scale(const float* __restrict__ diag,
                             float* __restrict__ dis, float* __restrict__ dsc) {
    int i = blockIdx.x * blockDim.x + threadIdx.x;
    if (i >= MROWS) return;
    float v = diag[i];
    float r = rsqrtf(v + 1.0f);
    dis[i] = r;
    dsc[i] = v * r * r;
}

// ---------------------------------------------------------------------------
// Kernel 4: xb = -( (W_left mix over D) @ W_right.T )  via V_WMMA_F32_16X16X4_F32
// Also seeds y = diag_scaled * xb.
// Block = 256 threads = 8 waves; W_right (128x128, 64KB) staged in LDS.
// Each wave: one 16-row strip of M, all 128 columns (8 accumulator tiles).
// A-fragment (16x4 f32, 2 VGPRs): lane holds A[lane%16, 2*(lane/16)+j];
// W_left 2x2 mix folded into the A load. B-fragment read straight from LDS
// (W_right row-major: B[k, col] = W_right[col, k] -> consecutive j => b64 load).
// ---------------------------------------------------------------------------
__global__ void __launch_bounds__(256)
k_gemm_wmma(const float* __restrict__ x, const float* __restrict__ W_left,
            const float* __restrict__ W_right, const float* __restrict__ dsc,
            float* __restrict__ xb, float* __restrict__ y) {
    __shared__ float sW[HCH * HCH];  // 64 KB
    int tid = threadIdx.x;
    for (int i = tid; i < (HCH * HCH) / 4; i += 256)
        ((v4f*)sW)[i] = ((const v4f*)W_right)[i];
    __syncthreads();

    int wave = tid >> 5, lane = tid & 31;
    int chunk = blockIdx.x * 8 + wave;       // 16-row strip index
    if (chunk * 16 >= MROWS) return;         // wave-uniform: EXEC stays all-ones
    int base = chunk * 16;

    int half = lane >> 4, mrow = lane & 15;
    int arow = base + mrow;                  // output row this lane feeds into A
    int node = arow >> 1, dc = arow & 1;
    float wl0 = W_left[dc * 2 + 0];
    float wl1 = W_left[dc * 2 + 1];
    const float* xr0 = x + (size_t)(node * 2 + 0) * HCH;
    const float* xr1 = x + (size_t)(node * 2 + 1) * HCH;

    v8f zacc = {};
    v8f acc[8];
#pragma unroll
    for (int ct = 0; ct < 8; ++ct) acc[ct] = zacc;

#pragma unroll 4
    for (int s = 0; s < 32; ++s) {
        int kb = 4 * s + 2 * half;           // K pair base for this lane
        v2f xa0 = *(const v2f*)(xr0 + kb);
        v2f xa1 = *(const v2f*)(xr1 + kb);
        v2f afrag;
        afrag[0] = wl0 * xa0[0] + wl1 * xa1[0];
        afrag[1] = wl0 * xa0[1] + wl1 * xa1[1];
#pragma unroll
        for (int ct = 0; ct < 8; ++ct) {
            v2f bfrag = *(const v2f*)(&sW[(ct * 16 + mrow) * HCH + kb]);
            acc[ct] = __builtin_amdgcn_wmma_f32_16x16x4_f32(
                false, afrag, false, bfrag, (short)0, acc[ct], false, false);
        }
    }

    // D layout: lane%16 = column, lanes 0-15 VGPR r -> M=r, lanes 16-31 -> M=8+r
#pragma unroll
    for (int r = 0; r < 8; ++r) {
        int orow = base + half * 8 + r;
        float ds = dsc[orow];
        size_t rb = (size_t)orow * HCH;
#pragma unroll
        for (int ct = 0; ct < 8; ++ct) {
            float v = -acc[ct][r];
            xb[rb + ct * 16 + mrow] = v;
            y[rb + ct * 16 + mrow] = ds * v;
        }
    }
}

// ---------------------------------------------------------------------------
// Kernel 5: edge aggregation. One wave per undirected edge; both directions.
// c[d'] = -p[i,d'] * dis[s,d'] * dis[d,d'] (symmetric under edge reversal).
// y[s] += c * xb[d];  y[d] += c * xb[s]   (elementwise over (d', h)).
// ---------------------------------------------------------------------------
__global__ void k_edge_agg(const int* __restrict__ src, const int* __restrict__ dst,
                           const float* __restrict__ p, const float* __restrict__ dis,
                           const float* __restrict__ xb, float* __restrict__ y) {
    int tid = blockIdx.x * blockDim.x + threadIdx.x;
    int e = tid >> 5;
    int lane = tid & 31;
    if (e >= NE0) return;
    int s = src[e], d = dst[e];
    float c0 = -p[2 * e + 0] * dis[2 * s + 0] * dis[2 * d + 0];
    float c1 = -p[2 * e + 1] * dis[2 * s + 1] * dis[2 * d + 1];
    size_t sb = (size_t)s * ROWLEN;
    size_t db = (size_t)d * ROWLEN;
    int off0 = 4 * lane;           // d'=0 region: [0,128)
    int off1 = 128 + 4 * lane;     // d'=1 region: [128,256)
    v4f vd0 = *(const v4f*)(xb + db + off0);
    v4f vs0 = *(const v4f*)(xb + sb + off0);
    v4f vd1 = *(const v4f*)(xb + db + off1);
    v4f vs1 = *(const v4f*)(xb + sb + off1);
#pragma unroll
    for (int q = 0; q < 4; ++q) {
        unsafeAtomicAdd(y + sb + off0 + q, c0 * vd0[q]);
        unsafeAtomicAdd(y + db + off0 + q, c0 * vs0[q]);
        unsafeAtomicAdd(y + sb + off1 + q, c1 * vd1[q]);
        unsafeAtomicAdd(y + db + off1 + q, c1 * vs1[q]);
    }
}

// ---------------------------------------------------------------------------
extern "C" void kernel_launch(void* const* d_in, const int* in_sizes, int n_in,
                              void* d_out, int out_size, void* d_ws, size_t ws_size,
                              hipStream_t stream) {
    // setup_inputs order: t, x, src, dst, W_sheaf, W_left, W_right
    const float* x       = (const float*)d_in[1];
    const int*   src     = (const int*)  d_in[2];
    const int*   dst     = (const int*)  d_in[3];
    const float* W_sheaf = (const float*)d_in[4];
    const float* W_left  = (const float*)d_in[5];
    const float* W_right = (const float*)d_in[6];
    float* y = (float*)d_out;

    float* ws   = (float*)d_ws;
    float* a    = ws;                    // 2*N
    float* b    = a    + 2 * N_NODES;    // 2*N
    float* diag = b    + 2 * N_NODES;    // 2*N
    float* dis  = diag + 2 * N_NODES;    // 2*N
    float* dsc  = dis  + 2 * N_NODES;    // 2*N
    float* p    = dsc  + 2 * N_NODES;    // 2*E0
    float* xb   = p    + 2 * NE0;        // N*D*HC (51.2 MB)

    k_zero<<<(2 * N_NODES + 255) / 256, 256, 0, stream>>>(diag, 2 * N_NODES);
    k_node_proj<<<(N_NODES + 7) / 8, 256, 0, stream>>>(x, W_sheaf, a, b);
    k_edge_maps<<<(NE0 + 255) / 256, 256, 0, stream>>>(src, dst, a, b, diag, p);
    k_node_scale<<<(MROWS + 255) / 256, 256, 0, stream>>>(diag, dis, dsc);
    k_gemm_wmma<<<(MROWS / 16 + 7) / 8, 256, 0, stream>>>(x, W_left, W_right, dsc, xb, y);
    k_edge_agg<<<(NE0 * 32 + 255) / 256, 256, 0, stream>>>(src, dst, p, dis, xb, y);
}